// SumPooling_54700703482382
// MI455X (gfx1250) — compile-verified
//
#include <hip/hip_runtime.h>

#ifndef __has_builtin
#define __has_builtin(x) 0
#endif

#if __has_builtin(__builtin_amdgcn_global_load_async_to_lds_b128)
#define USE_ASYNC_LDS 1
#else
#define USE_ASYNC_LDS 0
#endif

#define NW   8   // waves per block (wave32)
#define PIPE 8   // async pipeline depth per wave

typedef int v4i __attribute__((vector_size(16)));
typedef __attribute__((address_space(1))) v4i* gv4i_p;   // global
typedef __attribute__((address_space(3))) v4i* lv4i_p;   // LDS

#define WAIT_ASYNCCNT(n) asm volatile("s_wait_asynccnt %0" :: "n"(n) : "memory")
#define WAIT_DSCNT0()    asm volatile("s_wait_dscnt 0" ::: "memory")

// Segment-owned streaming sum: block g fully computes out[g][:].
// Deterministic: no atomics, fixed accumulation order.
__global__ __launch_bounds__(NW * 32) void segsum_kernel(
    const float* __restrict__ feats, const int* __restrict__ seg,
    float* __restrict__ out, int N)
{
#if USE_ASYNC_LDS
    __shared__ float4 stage[NW][PIPE][32];   // 32 KB staging
#endif
    __shared__ float4 partial[NW][32];       // 4 KB cross-wave reduce

    const int g    = blockIdx.x;
    const int lane = threadIdx.x & 31;
    const int w    = threadIdx.x >> 5;

    // lower_bound(g) / lower_bound(g+1) over sorted ids (uniform per block)
    int lo = 0, hi = N;
    while (lo < hi) { int m = (lo + hi) >> 1; if (seg[m] <  g) lo = m + 1; else hi = m; }
    const int s = lo;
    hi = N;
    while (lo < hi) { int m = (lo + hi) >> 1; if (seg[m] <= g) lo = m + 1; else hi = m; }
    const int e = lo;

    const int n  = e - s;
    const int Lw = (n > w) ? ((n - w + NW - 1) / NW) : 0;  // rows for this wave

    const float4* fv = (const float4*)feats;   // row r = fv[r*32 + lane]
    float4 acc; acc.x = acc.y = acc.z = acc.w = 0.0f;

#if USE_ASYNC_LDS
    // Prologue: fill the pipeline.
    const int pre = (Lw < PIPE) ? Lw : PIPE;
    for (int i = 0; i < pre; ++i) {
        const int r = s + w + i * NW;
        const float4* gp = fv + (size_t)r * 32 + lane;
        __builtin_amdgcn_global_load_async_to_lds_b128(
            (gv4i_p)gp, (lv4i_p)&stage[w][i][lane], 0, 0);
    }
    int main_iters = Lw - PIPE; if (main_iters < 0) main_iters = 0;
    // Steady state: wait oldest -> consume -> refill same slot.
    for (int i = 0; i < main_iters; ++i) {
        WAIT_ASYNCCNT(PIPE - 1);                       // oldest copy landed in LDS
        float4 v = stage[w][i & (PIPE - 1)][lane];
        acc.x += v.x; acc.y += v.y; acc.z += v.z; acc.w += v.w;
        WAIT_DSCNT0();                                 // slot fully read before reuse
        const int r = s + w + (i + PIPE) * NW;
        const float4* gp = fv + (size_t)r * 32 + lane;
        __builtin_amdgcn_global_load_async_to_lds_b128(
            (gv4i_p)gp, (lv4i_p)&stage[w][i & (PIPE - 1)][lane], 0, 0);
    }
    // Drain tail.
    WAIT_ASYNCCNT(0);
    for (int i = main_iters; i < Lw; ++i) {
        float4 v = stage[w][i & (PIPE - 1)][lane];
        acc.x += v.x; acc.y += v.y; acc.z += v.z; acc.w += v.w;
    }
#else
    for (int i = 0; i < Lw; ++i) {
        const int r = s + w + i * NW;
        float4 v = fv[(size_t)r * 32 + lane];
        acc.x += v.x; acc.y += v.y; acc.z += v.z; acc.w += v.w;
    }
#endif

    partial[w][lane] = acc;
    __syncthreads();

    if (w == 0) {
        float4 t = partial[0][lane];
#pragma unroll
        for (int k = 1; k < NW; ++k) {
            float4 p = partial[k][lane];
            t.x += p.x; t.y += p.y; t.z += p.z; t.w += p.w;
        }
        ((float4*)out)[(size_t)g * 32 + lane] = t;   // 128 floats per segment
    }
}

extern "C" void kernel_launch(void* const* d_in, const int* in_sizes, int n_in,
                              void* d_out, int out_size, void* d_ws, size_t ws_size,
                              hipStream_t stream) {
    const float* feats = (const float*)d_in[0];
    const int*   seg   = (const int*)d_in[1];
    float*       out   = (float*)d_out;
    const int D = 128;
    const int N = in_sizes[0] / D;       // 1048576
    const int G = out_size   / D;        // 1024
    hipLaunchKernelGGL(segsum_kernel, dim3(G), dim3(NW * 32), 0, stream,
                       feats, seg, out, N);
}